// MambaBlock_37263136260204
// MI455X (gfx1250) — compile-verified
//
#include <hip/hip_runtime.h>
#include <hip/hip_bf16.h>
#include <math.h>

#define D_MODEL 512
#define D_INNER 1024
#define D_STATE 64
#define D_CONV  4
#define DT_RANK 32
#define BB 2
#define LL 1000
#define NTOK (BB * LL)
#define EPS 1e-5f

typedef float v2f __attribute__((ext_vector_type(2)));
typedef float v4f __attribute__((ext_vector_type(4)));
typedef float v8f __attribute__((ext_vector_type(8)));

__device__ __forceinline__ float silu_f(float x) { return x / (1.0f + __expf(-x)); }

// ---------------------------------------------------------------------------
// RMSNorm: one block per token; writes forward h (f_norm_w) and time-flipped
// h_rev (b_norm_w) so the backward pipeline runs in flipped coordinates.
// ---------------------------------------------------------------------------
__global__ void mamba_rmsnorm_kernel(const float* __restrict__ x,
                                     const float* __restrict__ wf,
                                     const float* __restrict__ wb,
                                     float* __restrict__ h,
                                     float* __restrict__ h_rev) {
  int row = blockIdx.x;            // token in [0, NTOK)
  int tid = threadIdx.x;           // 256 threads, 2 elems each
  __shared__ float red[256];
  float v0 = x[row * D_MODEL + tid];
  float v1 = x[row * D_MODEL + tid + 256];
  red[tid] = v0 * v0 + v1 * v1;
  __syncthreads();
  for (int s = 128; s > 0; s >>= 1) {
    if (tid < s) red[tid] += red[tid + s];
    __syncthreads();
  }
  float inv = rsqrtf(red[0] * (1.0f / (float)D_MODEL) + EPS);
  int b = row / LL, t = row % LL;
  int rrow = b * LL + (LL - 1 - t);
  h[row * D_MODEL + tid]            = v0 * inv * wf[tid];
  h[row * D_MODEL + tid + 256]      = v1 * inv * wf[tid + 256];
  h_rev[rrow * D_MODEL + tid]       = v0 * inv * wb[tid];
  h_rev[rrow * D_MODEL + tid + 256] = v1 * inv * wb[tid + 256];
}

// ---------------------------------------------------------------------------
// FP32 WMMA GEMM: Y[M,N] = X[M,K] * W[N,K]^T  via V_WMMA_F32_16X16X4_F32.
// Register-blocked (16x64 strip, 4 accumulators per wave).  The K loop is
// processed in 8-wide chunks with a REMAPPED fragment layout: since WMMA
// accumulation is commutative over K and only requires A/B slot agreement,
// slot (half h, comp j) of step s holds memory k = k0 + 4h + 2s + j.  A
// lane's data for both steps of a chunk is then 4 consecutive floats at
// k0 + 4h  ->  ONE global_load_b128 per matrix per chunk: 5 b128 loads feed
// 8 WMMAs.  Chunks are double-buffered; the last chunk is peeled (no OOB).
// Ragged N strips clamp B row indices and discard at the store guard, so the
// loop is branch-free and EXEC stays all-ones around every WMMA.
// C/D layout per ISA 7.12.2: element v -> (M = v + 8*half, N = lane&15).
// Requires K % 8 == 0 and 16B-aligned rows (K, ld* multiples of 4 floats).
// ---------------------------------------------------------------------------
__global__ void wmma_gemm_f32_kernel(const float* __restrict__ X, int ldx,
                                     const float* __restrict__ W, int ldw,
                                     float* __restrict__ Y, int ldy,
                                     int M, int N, int K) {
  int wave = threadIdx.x >> 5;
  int lane = threadIdx.x & 31;
  int n0 = (blockIdx.x * 4 + wave) * 64;   // 64-wide N strip per wave
  int m0 = blockIdx.y * 16;
  if (n0 >= N || m0 >= M) return;          // wave-uniform guard
  int half = lane >> 4;
  int l15  = lane & 15;

  // Clamped W row indices: always in-bounds, so the K loop needs no guards.
  int r0 = min(n0 + l15,      N - 1);
  int r1 = min(n0 + 16 + l15, N - 1);
  int r2 = min(n0 + 32 + l15, N - 1);
  int r3 = min(n0 + 48 + l15, N - 1);

  const float* ap  = X + (size_t)(m0 + l15) * ldx + 4 * half;
  const float* bp0 = W + (size_t)r0 * ldw + 4 * half;
  const float* bp1 = W + (size_t)r1 * ldw + 4 * half;
  const float* bp2 = W + (size_t)r2 * ldw + 4 * half;
  const float* bp3 = W + (size_t)r3 * ldw + 4 * half;

  v8f acc0 = {}, acc1 = {}, acc2 = {}, acc3 = {};

  // Prologue: b128 fragments for chunk k0 = 0.
  v4f a4 = *(const v4f*)ap;
  v4f c0 = *(const v4f*)bp0;
  v4f c1 = *(const v4f*)bp1;
  v4f c2 = *(const v4f*)bp2;
  v4f c3 = *(const v4f*)bp3;

  // Pipelined main loop over 8-wide K chunks.
  for (int k0 = 0; k0 + 8 < K; k0 += 8) {
    int kn = k0 + 8;
    v4f a4n = *(const v4f*)(ap  + kn);
    v4f c0n = *(const v4f*)(bp0 + kn);
    v4f c1n = *(const v4f*)(bp1 + kn);
    v4f c2n = *(const v4f*)(bp2 + kn);
    v4f c3n = *(const v4f*)(bp3 + kn);
    __builtin_prefetch(ap  + kn + 64, 0, 1);   // global_prefetch_b8
    __builtin_prefetch(bp0 + kn + 64, 0, 1);

    v2f aA = {a4.x, a4.y}, aB = {a4.z, a4.w};
    v2f b0A = {c0.x, c0.y}, b0B = {c0.z, c0.w};
    v2f b1A = {c1.x, c1.y}, b1B = {c1.z, c1.w};
    v2f b2A = {c2.x, c2.y}, b2B = {c2.z, c2.w};
    v2f b3A = {c3.x, c3.y}, b3B = {c3.z, c3.w};
    acc0 = __builtin_amdgcn_wmma_f32_16x16x4_f32(false, aA, false, b0A,
                                                 (short)0, acc0, false, false);
    acc1 = __builtin_amdgcn_wmma_f32_16x16x4_f32(false, aA, false, b1A,
                                                 (short)0, acc1, false, false);
    acc2 = __builtin_amdgcn_wmma_f32_16x16x4_f32(false, aA, false, b2A,
                                                 (short)0, acc2, false, false);
    acc3 = __builtin_amdgcn_wmma_f32_16x16x4_f32(false, aA, false, b3A,
                                                 (short)0, acc3, false, false);
    acc0 = __builtin_amdgcn_wmma_f32_16x16x4_f32(false, aB, false, b0B,
                                                 (short)0, acc0, false, false);
    acc1 = __builtin_amdgcn_wmma_f32_16x16x4_f32(false, aB, false, b1B,
                                                 (short)0, acc1, false, false);
    acc2 = __builtin_amdgcn_wmma_f32_16x16x4_f32(false, aB, false, b2B,
                                                 (short)0, acc2, false, false);
    acc3 = __builtin_amdgcn_wmma_f32_16x16x4_f32(false, aB, false, b3B,
                                                 (short)0, acc3, false, false);
    a4 = a4n; c0 = c0n; c1 = c1n; c2 = c2n; c3 = c3n;
  }

  // Epilogue: last 8-wide chunk.
  {
    v2f aA = {a4.x, a4.y}, aB = {a4.z, a4.w};
    v2f b0A = {c0.x, c0.y}, b0B = {c0.z, c0.w};
    v2f b1A = {c1.x, c1.y}, b1B = {c1.z, c1.w};
    v2f b2A = {c2.x, c2.y}, b2B = {c2.z, c2.w};
    v2f b3A = {c3.x, c3.y}, b3B = {c3.z, c3.w};
    acc0 = __builtin_amdgcn_wmma_f32_16x16x4_f32(false, aA, false, b0A,
                                                 (short)0, acc0, false, false);
    acc1 = __builtin_amdgcn_wmma_f32_16x16x4_f32(false, aA, false, b1A,
                                                 (short)0, acc1, false, false);
    acc2 = __builtin_amdgcn_wmma_f32_16x16x4_f32(false, aA, false, b2A,
                                                 (short)0, acc2, false, false);
    acc3 = __builtin_amdgcn_wmma_f32_16x16x4_f32(false, aA, false, b3A,
                                                 (short)0, acc3, false, false);
    acc0 = __builtin_amdgcn_wmma_f32_16x16x4_f32(false, aB, false, b0B,
                                                 (short)0, acc0, false, false);
    acc1 = __builtin_amdgcn_wmma_f32_16x16x4_f32(false, aB, false, b1B,
                                                 (short)0, acc1, false, false);
    acc2 = __builtin_amdgcn_wmma_f32_16x16x4_f32(false, aB, false, b2B,
                                                 (short)0, acc2, false, false);
    acc3 = __builtin_amdgcn_wmma_f32_16x16x4_f32(false, aB, false, b3B,
                                                 (short)0, acc3, false, false);
  }

  float* yp = Y + (size_t)(m0 + 8 * half) * ldy + n0 + l15;
#pragma unroll
  for (int v = 0; v < 8; ++v) yp[(size_t)v * ldy] = acc0[v];
  if (n0 + 16 < N) {
#pragma unroll
    for (int v = 0; v < 8; ++v) yp[(size_t)v * ldy + 16] = acc1[v];
  }
  if (n0 + 32 < N) {
#pragma unroll
    for (int v = 0; v < 8; ++v) yp[(size_t)v * ldy + 32] = acc2[v];
  }
  if (n0 + 48 < N) {
#pragma unroll
    for (int v = 0; v < 8; ++v) yp[(size_t)v * ldy + 48] = acc3[v];
  }
}

// ---------------------------------------------------------------------------
// Causal depthwise conv (K=4) + SiLU.  xz holds [xc | z] per token (2*Di).
// ---------------------------------------------------------------------------
__global__ void dwconv_silu_kernel(const float* __restrict__ xz,
                                   const float* __restrict__ w,
                                   const float* __restrict__ bias,
                                   float* __restrict__ out) {
  int idx = blockIdx.x * blockDim.x + threadIdx.x;
  if (idx >= NTOK * D_INNER) return;
  int d = idx % D_INNER;
  int row = idx / D_INNER;
  int b = row / LL, t = row % LL;
  float acc = bias[d];
#pragma unroll
  for (int k = 0; k < D_CONV; ++k) {
    int tt = t - (D_CONV - 1) + k;
    if (tt >= 0)
      acc += w[d * D_CONV + k] * xz[(size_t)(b * LL + tt) * (2 * D_INNER) + d];
  }
  out[idx] = silu_f(acc);
}

// ---------------------------------------------------------------------------
// dt = softplus(dt_raw + dt_bias)
// ---------------------------------------------------------------------------
__global__ void dt_softplus_kernel(float* __restrict__ dt,
                                   const float* __restrict__ dtb) {
  int idx = blockIdx.x * blockDim.x + threadIdx.x;
  if (idx >= NTOK * D_INNER) return;
  int d = idx % D_INNER;
  float v = dt[idx] + dtb[d];
  dt[idx] = (v > 20.0f) ? v : log1pf(__expf(v));
}

// ---------------------------------------------------------------------------
// Selective scan: one wave32 per (b, d) channel; the 64-wide state vector is
// split 2 states per lane.  y_t = sum_n h_n * C_n via wave-wide shfl_xor.
// ---------------------------------------------------------------------------
__global__ void selective_scan_kernel(const float* __restrict__ u,     // xc (B,L,Di)
                                      const float* __restrict__ dt,    // (B,L,Di)
                                      const float* __restrict__ xdbl,  // (B,L,160): [dt|B|C]
                                      const float* __restrict__ A_log, // (Di,64)
                                      float* __restrict__ y) {         // (B,L,Di)
  int wid  = (blockIdx.x * blockDim.x + threadIdx.x) >> 5;
  int lane = threadIdx.x & 31;
  if (wid >= BB * D_INNER) return;
  int b = wid / D_INNER, d = wid % D_INNER;

  float a0 = -__expf(A_log[d * D_STATE + lane]);
  float a1 = -__expf(A_log[d * D_STATE + lane + 32]);
  float h0 = 0.0f, h1 = 0.0f;

  const float* dtp = dt + (size_t)b * LL * D_INNER + d;
  const float* up  = u  + (size_t)b * LL * D_INNER + d;
  const float* xp  = xdbl + (size_t)b * LL * (DT_RANK + 2 * D_STATE);
  float* yp = y + (size_t)b * LL * D_INNER + d;

  for (int t = 0; t < LL; ++t) {
    float dtt = dtp[(size_t)t * D_INNER];     // uniform across wave
    float ut  = up[(size_t)t * D_INNER];      // uniform across wave
    const float* xr = xp + t * (DT_RANK + 2 * D_STATE);
    float Bn0 = xr[DT_RANK + lane];
    float Bn1 = xr[DT_RANK + 32 + lane];
    float Cn0 = xr[DT_RANK + D_STATE + lane];
    float Cn1 = xr[DT_RANK + D_STATE + 32 + lane];
    float dBu = dtt * ut;
    h0 = __expf(dtt * a0) * h0 + dBu * Bn0;
    h1 = __expf(dtt * a1) * h1 + dBu * Bn1;
    float acc = h0 * Cn0 + h1 * Cn1;
#pragma unroll
    for (int off = 16; off > 0; off >>= 1) acc += __shfl_xor(acc, off, 32);
    if (lane == 0) yp[(size_t)t * D_INNER] = acc;
  }
}

// ---------------------------------------------------------------------------
// y = (y + xc * D) * silu(z)   (in-place on y)
// ---------------------------------------------------------------------------
__global__ void gate_combine_kernel(float* __restrict__ y,
                                    const float* __restrict__ xcv,
                                    const float* __restrict__ xz,
                                    const float* __restrict__ Dp) {
  int idx = blockIdx.x * blockDim.x + threadIdx.x;
  if (idx >= NTOK * D_INNER) return;
  int d = idx % D_INNER;
  int row = idx / D_INNER;
  float z = xz[(size_t)row * (2 * D_INNER) + D_INNER + d];
  y[idx] = (y[idx] + xcv[idx] * Dp[d]) * silu_f(z);
}

// ---------------------------------------------------------------------------
// out[t] = gemm_f[t] + gemm_b[L-1-t] + 2*x[t]
// (res_f = gemm_f + x ; flip(res_b) = flip(gemm_b) + x)
// ---------------------------------------------------------------------------
__global__ void merge_kernel(const float* __restrict__ rf,
                             const float* __restrict__ rb,
                             const float* __restrict__ x,
                             float* __restrict__ out) {
  int idx = blockIdx.x * blockDim.x + threadIdx.x;
  if (idx >= NTOK * D_MODEL) return;
  int row = idx / D_MODEL, e = idx % D_MODEL;
  int b = row / LL, t = row % LL;
  int rrow = b * LL + (LL - 1 - t);
  out[idx] = rf[idx] + rb[(size_t)rrow * D_MODEL + e] + 2.0f * x[idx];
}

// ---------------------------------------------------------------------------
extern "C" void kernel_launch(void* const* d_in, const int* in_sizes, int n_in,
                              void* d_out, int out_size, void* d_ws, size_t ws_size,
                              hipStream_t stream) {
  (void)in_sizes; (void)n_in; (void)out_size; (void)ws_size;
  const float* x = (const float*)d_in[0];
  // per-direction parameter base: f -> d_in[1..10], b -> d_in[11..20]
  // order: norm_w, in_proj, conv_w, conv_b, x_proj, dt_w, dt_b, A_log, D, out_proj
  const float* P[2][10];
  for (int p = 0; p < 2; ++p)
    for (int i = 0; i < 10; ++i) P[p][i] = (const float*)d_in[1 + p * 10 + i];

  float* ws = (float*)d_ws;
  size_t o = 0;
  float* h    = ws + o; o += (size_t)NTOK * D_MODEL;
  float* hrev = ws + o; o += (size_t)NTOK * D_MODEL;
  float *xz[2], *xcv[2], *xdbl[2], *dtb[2], *yb[2], *res[2];
  for (int p = 0; p < 2; ++p) {
    xz[p]   = ws + o; o += (size_t)NTOK * 2 * D_INNER;
    xcv[p]  = ws + o; o += (size_t)NTOK * D_INNER;
    xdbl[p] = ws + o; o += (size_t)NTOK * (DT_RANK + 2 * D_STATE);
    dtb[p]  = ws + o; o += (size_t)NTOK * D_INNER;
    yb[p]   = ws + o; o += (size_t)NTOK * D_INNER;
    res[p]  = ws + o; o += (size_t)NTOK * D_MODEL;
  }

  const int ew_blk = 256;
  const int n_di  = NTOK * D_INNER;
  const int mt    = NTOK / 16;          // 125 M-tiles
  auto nblk = [](int n) { return (n + 255) / 256; };  // N strips of 256 per block

  // Shared RMSNorm producing both direction inputs.
  mamba_rmsnorm_kernel<<<NTOK, 256, 0, stream>>>(x, P[0][0], P[1][0], h, hrev);

  for (int p = 0; p < 2; ++p) {
    const float* hin = (p == 0) ? h : hrev;
    // in_proj: (2000,512) x (2048,512)^T -> (2000,2048)
    wmma_gemm_f32_kernel<<<dim3(nblk(2 * D_INNER), mt), 128, 0, stream>>>(
        hin, D_MODEL, P[p][1], D_MODEL, xz[p], 2 * D_INNER,
        NTOK, 2 * D_INNER, D_MODEL);
    // causal dwconv + silu on xc half
    dwconv_silu_kernel<<<(n_di + ew_blk - 1) / ew_blk, ew_blk, 0, stream>>>(
        xz[p], P[p][2], P[p][3], xcv[p]);
    // x_proj: (2000,1024) x (160,1024)^T -> (2000,160)
    wmma_gemm_f32_kernel<<<dim3(nblk(DT_RANK + 2 * D_STATE), mt), 128, 0, stream>>>(
        xcv[p], D_INNER, P[p][4], D_INNER, xdbl[p], DT_RANK + 2 * D_STATE,
        NTOK, DT_RANK + 2 * D_STATE, D_INNER);
    // dt_proj: (2000,32) x (1024,32)^T -> (2000,1024)  (A strided in xdbl)
    wmma_gemm_f32_kernel<<<dim3(nblk(D_INNER), mt), 128, 0, stream>>>(
        xdbl[p], DT_RANK + 2 * D_STATE, P[p][5], DT_RANK, dtb[p], D_INNER,
        NTOK, D_INNER, DT_RANK);
    dt_softplus_kernel<<<(n_di + ew_blk - 1) / ew_blk, ew_blk, 0, stream>>>(
        dtb[p], P[p][6]);
    // selective scan: 2048 waves (one per (b,d) channel)
    selective_scan_kernel<<<(BB * D_INNER * 32) / 256, 256, 0, stream>>>(
        xcv[p], dtb[p], xdbl[p], P[p][7], yb[p]);
    // (y + xc*D) * silu(z)
    gate_combine_kernel<<<(n_di + ew_blk - 1) / ew_blk, ew_blk, 0, stream>>>(
        yb[p], xcv[p], xz[p], P[p][8]);
    // out_proj: (2000,1024) x (512,1024)^T -> (2000,512)
    wmma_gemm_f32_kernel<<<dim3(nblk(D_MODEL), mt), 128, 0, stream>>>(
        yb[p], D_INNER, P[p][9], D_INNER, res[p], D_MODEL,
        NTOK, D_MODEL, D_INNER);
  }

  int n_out = NTOK * D_MODEL;
  merge_kernel<<<(n_out + ew_blk - 1) / ew_blk, ew_blk, 0, stream>>>(
      res[0], res[1], x, (float*)d_out);
}